// VarRateResampler_14396730376489
// MI455X (gfx1250) — compile-verified
//
#include <hip/hip_runtime.h>
#include <math.h>

// VarRateResampler on MI455X (gfx1250).
//
// The lax.scan NCO has a closed form (output k has phase k*s - 128*t), so the
// per-(t,slot) sub-filter indices are computed in parallel -- no scan.
// Compute is a banded-tap-matrix matmul on V_WMMA_F32_16X16X4_F32:
//   D[16 batch rows x 16 (8 times x 2 slots)] += A[16x4]*B[4x16], K=28 total,
// with every element of each 16x16 D tile a useful output.
// Memory-bound (~100 MB moved -> ~4.3 us floor @ 23.3 TB/s; fits in 192MB L2).
// x is staged global->LDS with GLOBAL_LOAD_ASYNC_TO_LDS_B32 (ASYNCcnt path),
// transposed into a pitch-17 layout (coprime with 64 banks: conflict-free for
// both the strided async writes and the per-lane A-fragment ds reads).

typedef __attribute__((ext_vector_type(2))) float v2f;
typedef __attribute__((ext_vector_type(8))) float v8f;

#define QP        128
#define LTAPS     21
#define TLEN      4096
#define BROWS     2048
#define WG_TROWS  16            // batch rows per workgroup (WMMA M)
#define WG_TCOLS  64            // output times per workgroup
#define WAVE_TC   8             // output times per wave (8 t x 2 slots = N=16)
#define NQ        7             // K = 28 = 7 WMMA k-steps of 4
#define LDS_COLS  (WG_TCOLS + LTAPS - 1)   // 84 staged x columns
#define LDS_PITCH 17

__global__ __launch_bounds__(256)
void resample_wmma_kernel(const float* __restrict__ x,
                          const float* __restrict__ hs,
                          const float* __restrict__ accp,
                          float* __restrict__ out)
{
    __shared__ float xs[LDS_COLS * LDS_PITCH];

    const int t0  = blockIdx.x * WG_TCOLS;
    const int b0  = blockIdx.y * WG_TROWS;
    const int tid = threadIdx.x;

    // ---- Stage x[b0..b0+15, t0-20 .. t0+63] into LDS transposed [col][row]
    // via async DMA: per-lane global addr is coalesced along time; per-lane LDS
    // dest carries the transpose. Left edge (t<0) is zero-filled with ds writes.
    for (int i = tid; i < WG_TROWS * LDS_COLS; i += 256) {
        int r = i / LDS_COLS;
        int c = i - r * LDS_COLS;
        int tcol = t0 - (LTAPS - 1) + c;
        float* lp = &xs[c * LDS_PITCH + r];
        if (tcol >= 0) {
            const float* gp = &x[(size_t)(b0 + r) * TLEN + tcol];
            unsigned lds_off = (unsigned)(uintptr_t)lp;   // addr[31:0] = LDS offset
            asm volatile("global_load_async_to_lds_b32 %0, %1, off"
                         :: "v"(lds_off), "v"(gp) : "memory");
        } else {
            *lp = 0.0f;
        }
    }
#if __has_builtin(__builtin_amdgcn_s_wait_asynccnt)
    __builtin_amdgcn_s_wait_asynccnt(0);
#else
    asm volatile("s_wait_asynccnt 0x0" ::: "memory");
#endif
    __syncthreads();

    const int wave = tid >> 5;
    const int lane = tid & 31;
    const int m16  = lane & 15;
    const int t0w  = t0 + wave * WAVE_TC;     // first output time of this wave

    // ---- Per-lane NCO (closed form). Column n = lane%16 = (time tt, slot sl).
    const int n  = m16;
    const int tt = n >> 1;
    const int sl = n & 1;
    const int t  = t0w + tt;
    const double s = 64.0 + (double)accp[0];          // step_inc
    double k0   = ceil((128.0 * (double)t) / s);      // first output of step t
    double ind0 = k0 * s - 128.0 * (double)t;
    double ind  = (sl == 0) ? ind0 : (ind0 + s);
    bool valid  = (ind0 < 128.0) && (ind < 128.0);
    int row = -1;                                     // -1 -> NaN row (ref pads)
    if (valid) {
        int ri = (int)rint(ind);                      // round-half-even = jnp.round
        row = ri < (QP - 1) ? ri : (QP - 1);
    }
    const int row_c = (row >= 0) ? row : 0;
    const float* __restrict__ hrow = hs + row_c * LTAPS;

    const int koff = (lane < 16) ? 0 : 2;             // A/B per-lane K offset

    // ---- Build ALL B fragments up front, branchlessly (no EXEC churn):
    // unconditional clamped gathers from hs, then cndmask-select {tap, NaN, 0}.
    // Loads issue back-to-back, after the async staging has been waited on.
    float bvals[2 * NQ];
    #pragma unroll
    for (int q = 0; q < NQ; ++q) {
        #pragma unroll
        for (int j = 0; j < 2; ++j) {
            const int k = 4 * q + koff + j;
            const int l = k - tt;
            const bool in_band = ((unsigned)l < (unsigned)LTAPS);
            const float raw = hrow[in_band ? l : 0];  // always-executed load
            const float nz  = (row >= 0) ? raw : __builtin_nanf("");
            bvals[2 * q + j] = in_band ? nz : 0.0f;
        }
    }

    // ---- WMMA chain: D = sum_q A_q * B_q, A from LDS (ds_load_2addr pairs).
    v8f acc = {};
    #pragma unroll
    for (int q = 0; q < NQ; ++q) {
        const int col = wave * WAVE_TC + 4 * q + koff;
        v2f a;
        a.x = xs[col       * LDS_PITCH + m16];
        a.y = xs[(col + 1) * LDS_PITCH + m16];
        v2f b;
        b.x = bvals[2 * q];
        b.y = bvals[2 * q + 1];
        acc = __builtin_amdgcn_wmma_f32_16x16x4_f32(
                  /*neg_a=*/false, a, /*neg_b=*/false, b,
                  /*c_mod=*/(short)0, acc, /*reuse_a=*/false, /*reuse_b=*/false);
    }

    // ---- Store D. VGPR r holds M = r + (lane<16?0:8), N = lane%16; n maps
    // directly to the flat (t,slot) output offset -> each store writes two
    // contiguous 64B half-wave segments.
    const int mbase = (lane < 16) ? 0 : 8;
    #pragma unroll
    for (int r = 0; r < 8; ++r) {
        const int m = r + mbase;
        const size_t o = ((size_t)(b0 + m) * TLEN + t0w) * 2 + n;
        out[o] = acc[r];
    }
}

extern "C" void kernel_launch(void* const* d_in, const int* in_sizes, int n_in,
                              void* d_out, int out_size, void* d_ws, size_t ws_size,
                              hipStream_t stream) {
    const float* x    = (const float*)d_in[0];   // (2048, 4096) f32
    const float* hs   = (const float*)d_in[1];   // (128, 21)    f32
    const float* accp = (const float*)d_in[2];   // scalar       f32
    float* out = (float*)d_out;                  // (2048, 4096, 2) f32

    dim3 grid(TLEN / WG_TCOLS, BROWS / WG_TROWS);   // (64, 128)
    resample_wmma_kernel<<<grid, 256, 0, stream>>>(x, hs, accp, out);
}